// LocalAttentionCache_26774826123568
// MI455X (gfx1250) — compile-verified
//
#include <hip/hip_runtime.h>
#include <math.h>
#include <stdint.h>

typedef __attribute__((ext_vector_type(2))) float v2f;
typedef __attribute__((ext_vector_type(8))) float v8f;

#define NUM_FREQ 16
#define KNN      16
#define LSEQ     4096
#define NB       4
#define PE_DIM   64          // 4 * NUM_FREQ
#define PI_F     3.14159265358979323846f

// One wave (32 lanes) handles a 16-query tile. Lanes n and n+16 both own
// query n (WMMA 16x16 f32 C-layout: lanes 0-15 hold rows M=0..7, lanes 16-31
// hold rows M=8..15, column N = lane&15). Each lane keeps a private top-16
// over its half of the candidates; the two halves merge via shuffles.
__global__ __launch_bounds__(256) void knn_rpe_kernel(
    const float* __restrict__ positions,  // [NB, LSEQ, 2]
    float* __restrict__ out)
{
    __shared__ float posS[LSEQ * 2];  // 32 KB
    __shared__ float nrmS[LSEQ];      // 16 KB

    const int tid  = threadIdx.x;
    const int wave = tid >> 5;
    const int lane = tid & 31;
    const int hi   = lane >> 4;      // 0: rows 0-7 of tile, 1: rows 8-15
    const int n    = lane & 15;      // query column within tile

    const int b      = blockIdx.x >> 5;          // batch (4)
    const int qtile0 = (blockIdx.x & 31) * 128;  // 128 queries per block (8 waves)
    const float* bpos = positions + (size_t)b * LSEQ * 2;

    // ---- stage positions in LDS via CDNA5 async direct-to-LDS loads ----
    // 32 KB = 2048 x b128 transfers; each of 256 threads issues 8.
    {
        const unsigned ldsBase = (unsigned)(uintptr_t)posS;
        for (int i = tid; i < (LSEQ * 2) / 4; i += 256) {
            unsigned dst = ldsBase + (unsigned)i * 16u;
            const float4* g = ((const float4*)bpos) + i;
            asm volatile("global_load_async_to_lds_b128 %0, %1, off"
                         :: "v"(dst), "v"(g)
                         : "memory");
        }
        asm volatile("s_wait_asynccnt 0" ::: "memory");
    }
    __syncthreads();
    for (int i = tid; i < LSEQ; i += 256) {
        float x = posS[2 * i], y = posS[2 * i + 1];
        nrmS[i] = x * x + y * y;
    }
    __syncthreads();

    // ---- this lane's query ----
    const int   ql = qtile0 + wave * 16 + n;     // query index within batch
    const float qx = posS[2 * ql];
    const float qy = posS[2 * ql + 1];
    const float nq = qx * qx + qy * qy;

    // B operand (4x16 f32): K=0 row = query x, K=1 row = query y in lanes 0-15;
    // K=2,3 rows (lanes 16-31) zeroed.
    v2f bmat;
    bmat.x = hi ? 0.0f : qx;
    bmat.y = hi ? 0.0f : qy;

    // ---- private top-16 (registers only; unrolled cndmask updates) ----
    float bestd[KNN];
    int   besti[KNN];
#pragma unroll
    for (int j = 0; j < KNN; ++j) { bestd[j] = INFINITY; besti[j] = 0; }
    float worst = INFINITY;
    int   wslot = 0;

    // ---- sweep all 256 candidate tiles with WMMA f32 16x16x4 ----
    for (int mt = 0; mt < LSEQ / 16; ++mt) {
        const int mrow = mt * 16 + (lane & 15);
        v2f amat;
        {
            float ax = posS[2 * mrow];
            float ay = posS[2 * mrow + 1];
            amat.x = hi ? 0.0f : ax;   // lanes 0-15: K=0,1; lanes 16-31 (K=2,3) zero
            amat.y = hi ? 0.0f : ay;
        }

        v8f c = {};
        // D[m][n] = x_m*x_n + y_m*y_n  (dot products of candidate/query tiles)
        v8f d = __builtin_amdgcn_wmma_f32_16x16x4_f32(
            false, amat, false, bmat, (short)0, c, false, false);

        const int mbase = mt * 16 + 8 * hi;  // 32B-aligned -> two ds_load_b128
        float4 nm0 = *(const float4*)&nrmS[mbase];
        float4 nm1 = *(const float4*)&nrmS[mbase + 4];
        float nm[8] = {nm0.x, nm0.y, nm0.z, nm0.w, nm1.x, nm1.y, nm1.z, nm1.w};

#pragma unroll
        for (int r = 0; r < 8; ++r) {
            const int m = mbase + r;
            float dsq = nm[r] + nq - 2.0f * d[r];
            if (m == ql) dsq = INFINITY;  // mask diagonal
            if (dsq < worst) {
#pragma unroll
                for (int j = 0; j < KNN; ++j)
                    if (j == wslot) { bestd[j] = dsq; besti[j] = m; }
                worst = bestd[0]; wslot = 0;
#pragma unroll
                for (int j = 1; j < KNN; ++j)
                    if (bestd[j] > worst) { worst = bestd[j]; wslot = j; }
            }
        }
    }

    // ---- merge lane n+16's partial list into lane n via shuffles ----
    float od[KNN];
    int   oi[KNN];
#pragma unroll
    for (int j = 0; j < KNN; ++j) {
        od[j] = __shfl(bestd[j], (lane + 16) & 31, 32);
        oi[j] = __shfl(besti[j], (lane + 16) & 31, 32);
    }

    if (!hi) {
        float cv[32];
        int   ci[32];
#pragma unroll
        for (int j = 0; j < KNN; ++j) {
            cv[j] = bestd[j];      ci[j] = besti[j];
            cv[KNN + j] = od[j];   ci[KNN + j] = oi[j];
        }

        float outd[KNN];
        int   outi[KNN];
#pragma unroll
        for (int s = 0; s < KNN; ++s) {       // ascending selection (top_k order)
            float mv = cv[0];
            int   mi = 0;
#pragma unroll
            for (int t = 1; t < 32; ++t)
                if (cv[t] < mv) { mv = cv[t]; mi = t; }
            int midx = ci[0];
#pragma unroll
            for (int t = 0; t < 32; ++t)
                if (t == mi) { midx = ci[t]; cv[t] = INFINITY; }
            outd[s] = mv;
            outi[s] = midx;
        }
        (void)outd;

        // ---- emit outputs (tuple concatenated flat, float32) ----
        const size_t base = (size_t)b * LSEQ + (size_t)ql;
        const size_t O0 = 0;                                        // topk_indices
        const size_t O1 = O0 + (size_t)NB * LSEQ * KNN;             // rpe
        const size_t O2 = O1 + (size_t)NB * LSEQ * KNN * PE_DIM;    // self_rpe
        const size_t O3 = O2 + (size_t)NB * LSEQ * PE_DIM;          // distances
        const size_t O4 = O3 + (size_t)NB * LSEQ * KNN;             // neighbor_positions

        const float ps = 3.0f * __fsqrt_rn((float)KNN / PI_F);      // physical scale

        // self_rpe: sin(0)=0, cos(0)=1 blocks
#pragma unroll
        for (int f = 0; f < NUM_FREQ; ++f) {
            const size_t sb = O2 + base * PE_DIM;
            out[sb + f]                 = 0.0f;
            out[sb + NUM_FREQ + f]      = 1.0f;
            out[sb + 2 * NUM_FREQ + f]  = 0.0f;
            out[sb + 3 * NUM_FREQ + f]  = 1.0f;
        }

        for (int j = 0; j < KNN; ++j) {
            const int m  = outi[j];
            const float nx = posS[2 * m];
            const float ny = posS[2 * m + 1];
            const float dx = nx - qx;
            const float dy = ny - qy;
            const float dist = __fsqrt_rn(dx * dx + dy * dy + 1e-8f);

            out[O0 + base * KNN + j]            = (float)m;
            out[O3 + base * KNN + j]            = dist;
            out[O4 + (base * KNN + j) * 2 + 0]  = nx;
            out[O4 + (base * KNN + j) * 2 + 1]  = ny;

            const float sx = dx / ps;
            const float sy = dy / ps;
            const size_t rb = O1 + (base * KNN + (size_t)j) * PE_DIM;
            float fr = PI_F;
#pragma unroll
            for (int f = 0; f < NUM_FREQ; ++f) {
                const float axv = sx * fr;
                const float ayv = sy * fr;
                out[rb + f]                = __sinf(axv);
                out[rb + NUM_FREQ + f]     = __cosf(axv);
                out[rb + 2 * NUM_FREQ + f] = __sinf(ayv);
                out[rb + 3 * NUM_FREQ + f] = __cosf(ayv);
                fr *= 2.0f;
            }
        }
    }
}

extern "C" void kernel_launch(void* const* d_in, const int* in_sizes, int n_in,
                              void* d_out, int out_size, void* d_ws, size_t ws_size,
                              hipStream_t stream) {
    (void)in_sizes; (void)n_in; (void)d_ws; (void)ws_size; (void)out_size;
    const float* positions = (const float*)d_in[0];   // [4, 4096, 2] f32
    float* out = (float*)d_out;                       // concatenated tuple, f32

    // 4 batches * 32 query-blocks of 128 queries (8 waves * 16) each
    dim3 grid(NB * (LSEQ / 128));
    dim3 block(256);
    knn_rpe_kernel<<<grid, block, 0, stream>>>(positions, out);
}